// PygGCN_13365938225231
// MI455X (gfx1250) — compile-verified
//
#include <hip/hip_runtime.h>

#define N_NODES 50000
#define DIN 512
#define DH 128
#define DOUT 64

typedef __attribute__((ext_vector_type(2))) float v2f;
typedef __attribute__((ext_vector_type(8))) float v8f;

// ---------------- degree / normalization ----------------

__global__ void k_init_deg(float* __restrict__ deg) {
    int i = blockIdx.x * blockDim.x + threadIdx.x;
    if (i < N_NODES) deg[i] = 1.0f;   // self loop contributes 1
}

__global__ void k_count_deg(const long long* __restrict__ dst, float* __restrict__ deg, int E) {
    int e = blockIdx.x * blockDim.x + threadIdx.x;
    if (e < E) unsafeAtomicAdd(&deg[(int)dst[e]], 1.0f);
}

__global__ void k_dinv(const float* __restrict__ deg, float* __restrict__ dinv) {
    int i = blockIdx.x * blockDim.x + threadIdx.x;
    if (i < N_NODES) dinv[i] = rsqrtf(deg[i]);   // deg >= 1 always (self loops)
}

// ---- f32 WMMA GEMM with fused self-loop epilogue ----
// Y[N,M] = X[N,K] * W[K,M]        (raw transform, needed as scatter source)
// A[N,M] = Y * dinv[row]^2 + bias (self-loop contribution + bias, scatter target)
//
// One wave computes a 16 x (16*NT) slab: NT independent accumulators -> NT
// independent v_wmma_f32_16x16x4_f32 chains per k-step, A regs reused NT times.
// A layout (16x4 f32): lanes 0-15 -> row=lane, K={k,k+1}; lanes 16-31 -> K={k+2,k+3}.
// B layout (4x16 f32): lanes 0-15 -> col=lane, K={k,k+1}; lanes 16-31 -> K={k+2,k+3}.
// C/D: VGPR r -> row = r + 8*(lane>=16), col = lane&15.
template<int K, int M, int NT>
__global__ void k_gemm_wmma_fused(const float* __restrict__ X,
                                  const float* __restrict__ W,
                                  const float* __restrict__ dinv,
                                  const float* __restrict__ bias,
                                  float* __restrict__ Y,
                                  float* __restrict__ A) {
    constexpr int STRIPS = M / (16 * NT);
    constexpr int TOTAL  = (N_NODES / 16) * STRIPS;

    const int wid = blockIdx.x * (blockDim.x >> 5) + (threadIdx.x >> 5);
    if (wid >= TOTAL) return;                 // wave-uniform: EXEC stays all-ones

    const int lane   = threadIdx.x & 31;
    const int strip  = wid % STRIPS;
    const int tile_m = wid / STRIPS;
    const int half   = lane >> 4;             // 0 or 1
    const int l16    = lane & 15;
    const int koff   = half * 2;
    const int colbase = strip * (16 * NT) + l16;

    const float* xp = X + (size_t)(tile_m * 16 + l16) * K + koff;
    const float* wp = W + (size_t)koff * M + colbase;

    v8f c[NT];
#pragma unroll
    for (int t = 0; t < NT; t++) c[t] = (v8f){};

#pragma unroll 4
    for (int k = 0; k < K; k += 4) {
        v2f a;
        a.x = xp[0];  a.y = xp[1];            // one b64 load feeds NT WMMAs
#pragma unroll
        for (int t = 0; t < NT; t++) {
            v2f b;
            b.x = wp[t * 16];
            b.y = wp[t * 16 + M];
            c[t] = __builtin_amdgcn_wmma_f32_16x16x4_f32(
                       false, a, false, b, (short)0, c[t], false, false);
        }
        xp += 4;
        wp += 4 * M;
    }

    // fused epilogue: write raw transform + self-loop/bias-initialized aggregate
    float bias_v[NT];
#pragma unroll
    for (int t = 0; t < NT; t++) bias_v[t] = bias[colbase + t * 16];

    const int rowbase = tile_m * 16 + half * 8;
#pragma unroll
    for (int r = 0; r < 8; r++) {
        float dv  = dinv[rowbase + r];
        float dv2 = dv * dv;
        size_t rowoff = (size_t)(rowbase + r) * M + colbase;
#pragma unroll
        for (int t = 0; t < NT; t++) {
            float val = c[t][r];
            Y[rowoff + t * 16] = val;
            A[rowoff + t * 16] = fmaf(val, dv2, bias_v[t]);
        }
    }
}

// ---------------- edge scatter: A[dst,:] += T[src,:] * dinv[src]*dinv[dst] ----------------
// One wave per edge; lane covers F/32 contiguous floats (f32 atomics resolve in L2:
// aggregate buffers are 12.8-25.6 MB, resident in the 192 MB L2).
template<int F>
__global__ void k_scatter(const float* __restrict__ T,
                          const long long* __restrict__ src,
                          const long long* __restrict__ dst,
                          const float* __restrict__ dinv,
                          float* __restrict__ A, int E) {
    int wid = blockIdx.x * (blockDim.x >> 5) + (threadIdx.x >> 5);
    if (wid >= E) return;
    int lane = threadIdx.x & 31;
    int s = (int)src[wid], d = (int)dst[wid];
    float nrm = dinv[s] * dinv[d];
    constexpr int PER = F / 32;   // 4 (F=128) or 2 (F=64)
    const float* ip = T + (size_t)s * F + lane * PER;
    float*       op = A + (size_t)d * F + lane * PER;
    float v[PER];
#pragma unroll
    for (int j = 0; j < PER; j++) v[j] = ip[j];
#pragma unroll
    for (int j = 0; j < PER; j++) unsafeAtomicAdd(op + j, v[j] * nrm);
}

__global__ void k_relu(float* __restrict__ A, int n) {
    int idx = blockIdx.x * blockDim.x + threadIdx.x;
    if (idx < n) A[idx] = fmaxf(A[idx], 0.0f);
}

// ---------------- host launch ----------------

extern "C" void kernel_launch(void* const* d_in, const int* in_sizes, int n_in,
                              void* d_out, int out_size, void* d_ws, size_t ws_size,
                              hipStream_t stream) {
    const float*     x   = (const float*)d_in[0];
    const long long* ei  = (const long long*)d_in[1];   // int64 [2, E]
    const float*     W1  = (const float*)d_in[2];
    const float*     b1  = (const float*)d_in[3];
    const float*     W2  = (const float*)d_in[4];
    const float*     b2  = (const float*)d_in[5];
    float*           out = (float*)d_out;

    const int E = in_sizes[1] / 2;                       // 800000
    const long long* src = ei;
    const long long* dst = ei + E;

    // workspace layout (floats)
    float* T1   = (float*)d_ws;                 // [N, DH]   x@W1 (scatter source)
    float* A1   = T1 + (size_t)N_NODES * DH;    // [N, DH]   aggregate -> H1 after relu
    float* T2   = A1 + (size_t)N_NODES * DH;    // [N, DOUT] h1@W2 (scatter source)
    float* deg  = T2 + (size_t)N_NODES * DOUT;  // [N]
    float* dinv = deg + N_NODES;                // [N]

    const int TPB = 256;
    const int nBlkN   = (N_NODES + TPB - 1) / TPB;
    const int nBlkE   = (E + TPB - 1) / TPB;
    const int nBlkNDH = (N_NODES * DH + TPB - 1) / TPB;
    const int nBlkEdgeWave = (E + 7) / 8;       // 8 waves/block, 1 wave/edge

    // normalization
    k_init_deg<<<nBlkN, TPB, 0, stream>>>(deg);
    k_count_deg<<<nBlkE, TPB, 0, stream>>>(dst, deg, E);
    k_dinv<<<nBlkN, TPB, 0, stream>>>(deg, dinv);

    // layer 1: T1 = x@W1, A1 = T1*dinv^2 + b1   (3125 m-tiles * 2 strips = 6250 waves)
    {
        constexpr int waves = (N_NODES / 16) * (DH / 64);
        k_gemm_wmma_fused<DIN, DH, 4><<<(waves + 7) / 8, 256, 0, stream>>>(
            x, W1, dinv, b1, T1, A1);
    }
    k_scatter<DH><<<nBlkEdgeWave, 256, 0, stream>>>(T1, src, dst, dinv, A1, E);
    k_relu<<<nBlkNDH, TPB, 0, stream>>>(A1, N_NODES * DH);

    // layer 2: T2 = H1@W2, out = T2*dinv^2 + b2  (3125 waves)
    {
        constexpr int waves = (N_NODES / 16) * (DOUT / 64);
        k_gemm_wmma_fused<DH, DOUT, 4><<<(waves + 7) / 8, 256, 0, stream>>>(
            A1, W2, dinv, b2, T2, out);
    }
    k_scatter<DOUT><<<nBlkEdgeWave, 256, 0, stream>>>(T2, src, dst, dinv, out, E);
}